// REModel_54717883351135
// MI455X (gfx1250) — compile-verified
//
#include <hip/hip_runtime.h>
#include <math.h>

// Problem constants (match reference)
#define BB 64
#define SS 2048
#define HH 1024
#define LL 15

typedef __attribute__((ext_vector_type(2))) float v2f;
typedef __attribute__((ext_vector_type(8))) float v8f;

// ---------------------------------------------------------------------------
// Kernel 1: pooling + tanh.  A[b, 0, h] = tanh(x[b,0,h])
//                            A[b, 1, h] = tanh(mean_{s in [s1,e1)} x[b,s,h])
//                            A[b, 2, h] = tanh(mean_{s in [s2,e2)} x[b,s,h])
// Coalesced on h; spans are <= 31 rows so this is tiny traffic.
// ---------------------------------------------------------------------------
__global__ void pool_tanh_kernel(const float* __restrict__ x,
                                 const int* __restrict__ eidx,
                                 float* __restrict__ A) {
  const int b = blockIdx.x;
  const int s1 = eidx[b * 4 + 0], e1 = eidx[b * 4 + 1];
  const int s2 = eidx[b * 4 + 2], e2 = eidx[b * 4 + 3];
  const float inv1 = 1.0f / fmaxf((float)(e1 - s1), 1.0f);
  const float inv2 = 1.0f / fmaxf((float)(e2 - s2), 1.0f);
  const float* xb = x + (size_t)b * SS * HH;
  float* Ab = A + (size_t)b * 3 * HH;
  for (int h = threadIdx.x; h < HH; h += blockDim.x) {
    // CLS row
    Ab[h] = tanhf(xb[h]);
    // subject span mean
    float acc1 = 0.0f;
    for (int s = s1; s < e1; ++s) acc1 += xb[(size_t)s * HH + h];
    Ab[HH + h] = tanhf(acc1 * inv1);
    // object span mean
    float acc2 = 0.0f;
    for (int s = s2; s < e2; ++s) acc2 += xb[(size_t)s * HH + h];
    Ab[2 * HH + h] = tanhf(acc2 * inv2);
  }
}

// ---------------------------------------------------------------------------
// Generic f32 WMMA GEMM:  C[:, seg*HH + n] = A_seg[64 x K] @ Bw[K x HH] + bias
//   - block = 512 threads = 16 waves; wave (my, nx) owns a 16x16 tile; the
//     block covers all M=64 rows so each weight element is pulled from HBM
//     exactly once per kernel.
//   - K / LDA / LDC are template constants and the weight row stride is HH,
//     so inner-loop addressing reduces to pointer bumps + immediate offsets.
//   - seg = blockIdx.y selects {B0,B1,B2} (3-way fused layer-1); A_seg is the
//     K-column slab starting at column seg*K of A.
//   - Per-lane fragment layout follows ISA 7.12.2 (32-bit A 16x4 / B 4x16 /
//     C 16x16): lane = 16*half + lm;  A: a[j] = A[m0+lm][k + 2*half + j];
//     B: b[j] = B[k + 2*half + j][n0+lm];  C: c[i] = C[m0 + 8*half + i][n0+lm].
// ---------------------------------------------------------------------------
template <int K, int LDA, int LDC>
__global__ __launch_bounds__(512)
void gemm_wmma_f32(const float* __restrict__ A,
                   const float* __restrict__ B0, const float* __restrict__ B1,
                   const float* __restrict__ B2,
                   const float* __restrict__ bias0, const float* __restrict__ bias1,
                   const float* __restrict__ bias2,
                   float* __restrict__ C) {
  const int seg = blockIdx.y;
  const float* __restrict__ Bw   = (seg == 0) ? B0    : ((seg == 1) ? B1    : B2);
  const float* __restrict__ bias = (seg == 0) ? bias0 : ((seg == 1) ? bias1 : bias2);

  const int wave = threadIdx.x >> 5;
  const int lane = threadIdx.x & 31;
  const int my = wave & 3;   // m-tile: 0..3  (M = 64)
  const int nx = wave >> 2;  // n-tile: 0..3  (block covers 64 columns)
  const int m0 = my * 16;
  const int n0 = blockIdx.x * 64 + nx * 16;
  const int half = lane >> 4;  // 0: K pair {0,1} / rows 0-7; 1: K pair {2,3} / rows 8-15
  const int lm = lane & 15;

  // A fragment pointer: row (m0+lm), starting column seg*K + 2*half.
  // Always 8-byte aligned (row*LDA + seg*K + k + 2*half is even), so the two
  // K-adjacent elements load as one global_load_b64.
  const float* ap = A + (size_t)(m0 + lm) * LDA + (size_t)seg * K + 2 * half;
  // B fragment pointer: column (n0+lm), starting row 2*half. Row stride = HH.
  const float* bp = Bw + (size_t)(2 * half) * HH + (n0 + lm);

  v8f acc = {};
#pragma unroll 8
  for (int k = 0; k < K; k += 4) {
    v2f a = *(const v2f*)ap;            // A[m][ka], A[m][ka+1]
    v2f bfrag;
    bfrag.x = bp[0];                    // B[ka][n]
    bfrag.y = bp[HH];                   // B[ka+1][n]
    // D = A(16x4, f32) * B(4x16, f32) + C  -> v_wmma_f32_16x16x4_f32
    acc = __builtin_amdgcn_wmma_f32_16x16x4_f32(
        false, a, false, bfrag, (short)0, acc, false, false);
    ap += 4;
    bp += 4 * HH;
  }

  const int gn = seg * HH + n0 + lm;
  const float bv = bias[n0 + lm];
  const int mrow = m0 + half * 8;
#pragma unroll
  for (int i = 0; i < 8; ++i) {
    C[(size_t)(mrow + i) * LDC + gn] = acc[i] + bv;
  }
}

// ---------------------------------------------------------------------------
// Kernel 5: out[b, l] = h3[b,:] . Wout[:, l] + bout[l]   (64 x 15, K = 1024)
// 15 waves per block (one per label), wave32 shuffle reduction.
// ---------------------------------------------------------------------------
__global__ void head_kernel(const float* __restrict__ h3,
                            const float* __restrict__ Wout,
                            const float* __restrict__ bout,
                            float* __restrict__ out) {
  const int b = blockIdx.x;
  const int w = threadIdx.x >> 5;  // label index 0..14
  const int lane = threadIdx.x & 31;
  const float* hb = h3 + (size_t)b * HH;
  float sum = 0.0f;
  for (int k = lane; k < HH; k += 32) {
    sum += hb[k] * Wout[(size_t)k * LL + w];
  }
#pragma unroll
  for (int off = 16; off >= 1; off >>= 1) {
    sum += __shfl_xor(sum, off, 32);
  }
  if (lane == 0) out[b * LL + w] = sum + bout[w];
}

// ---------------------------------------------------------------------------
// Launcher
// ---------------------------------------------------------------------------
extern "C" void kernel_launch(void* const* d_in, const int* in_sizes, int n_in,
                              void* d_out, int out_size, void* d_ws, size_t ws_size,
                              hipStream_t stream) {
  const float* x     = (const float*)d_in[0];
  const int*   eidx  = (const int*)d_in[1];
  const float* W_cls = (const float*)d_in[2];
  const float* b_cls = (const float*)d_in[3];
  const float* W_e1  = (const float*)d_in[4];
  const float* b_e1  = (const float*)d_in[5];
  const float* W_e2  = (const float*)d_in[6];
  const float* b_e2  = (const float*)d_in[7];
  const float* W1    = (const float*)d_in[8];
  const float* b1    = (const float*)d_in[9];
  const float* W2    = (const float*)d_in[10];
  const float* b2    = (const float*)d_in[11];
  const float* Wout  = (const float*)d_in[12];
  const float* bout  = (const float*)d_in[13];
  float* out = (float*)d_out;

  // Scratch layout (floats): A[64*3072] | H1[64*3072] | h2[64*1024] | h3[64*1024]
  float* A  = (float*)d_ws;
  float* H1 = A + (size_t)BB * 3 * HH;
  float* h2 = H1 + (size_t)BB * 3 * HH;
  float* h3 = h2 + (size_t)BB * HH;

  // 1) pooling + tanh -> A [64, 3, 1024]
  pool_tanh_kernel<<<BB, 256, 0, stream>>>(x, eidx, A);

  // 2) fused layer-1: H1[:, seg*1024:+1024] = A[:,seg,:] @ {W_cls,W_e1,W_e2} + bias
  gemm_wmma_f32<HH, 3 * HH, 3 * HH><<<dim3(HH / 64, 3), 512, 0, stream>>>(
      A, W_cls, W_e1, W_e2, b_cls, b_e1, b_e2, H1);

  // 3) h2 = H1 @ W1 + b1   (K = 3072)
  gemm_wmma_f32<3 * HH, 3 * HH, HH><<<dim3(HH / 64, 1), 512, 0, stream>>>(
      H1, W1, W1, W1, b1, b1, b1, h2);

  // 4) h3 = h2 @ W2 + b2   (K = 1024)
  gemm_wmma_f32<HH, HH, HH><<<dim3(HH / 64, 1), 512, 0, stream>>>(
      h2, W2, W2, W2, b2, b2, b2, h3);

  // 5) out = h3 @ Wout + bout  -> [64, 15]
  head_kernel<<<BB, LL * 32, 0, stream>>>(h3, Wout, bout, out);
}